// RetinaFocalLoss_71588514890242
// MI455X (gfx1250) — compile-verified
//
#include <hip/hip_runtime.h>

#define BATCH   16
#define NP      22536
#define NOBJ    32
#define NCLS    80
#define THRESH  0.5f
#define ALPHA_F 0.25f

typedef __attribute__((ext_vector_type(2))) float v2f;
typedef __attribute__((ext_vector_type(8))) float v8f;

// ---------------------------------------------------------------------------
// Exact wave32 sum reduction through the matrix pipe:
// A (16x4 f32): lane L's a[0] is A[L%16, (L<16)?0:2]; a[1] (cols 1,3) = 0.
// B = all-ones 4x16  =>  D[m,n] = x_m + x_{m+16}.
// Each lane sums its 8 D registers (rows m..m+7 of its column); lanes 0 and 16
// hold the two complementary half-sums of column 0 -> two atomicAdds = total.
// ---------------------------------------------------------------------------
__device__ __forceinline__ void wave_sum_wmma_atomic(float x, float* dst) {
    v2f a; a[0] = x;    a[1] = 0.0f;
    v2f b; b[0] = 1.0f; b[1] = 1.0f;
    v8f c = {};
    c = __builtin_amdgcn_wmma_f32_16x16x4_f32(
            /*neg_a=*/false, a, /*neg_b=*/false, b,
            /*c_mod=*/(short)0, c, /*reuse_a=*/false, /*reuse_b=*/false);
    float r = c[0] + c[1] + c[2] + c[3] + c[4] + c[5] + c[6] + c[7];
    if ((threadIdx.x & 15u) == 0u)   // lanes 0 and 16 of each wave
        atomicAdd(dst, r);
}

// ---------------------------------------------------------------------------
// Init: zero per-object best keys and the 4 global accumulators (every call).
// ---------------------------------------------------------------------------
__global__ void k_init(unsigned long long* __restrict__ packed,
                       float* __restrict__ accum) {
    unsigned i = threadIdx.x;
    if (i < BATCH * NOBJ) packed[i] = 0ull;
    if (i < 4) accum[i] = 0.0f;
}

// ---------------------------------------------------------------------------
// Matching: IoU of every prior vs 32 objects. Per-prior max/argmax written
// directly; per-object max/argmax via block-local scan + packed u64 atomicMax
// (key = iou_bits<<32 | ~prior  -> ties resolve to the LOWEST prior index).
// ---------------------------------------------------------------------------
__global__ void k_match(const float4* __restrict__ boxes,   // [B*32] xyxy
                        const float4* __restrict__ priors,  // [NP]  cxcywh
                        float* __restrict__ ov_out,         // [B*NP]
                        int* __restrict__ obj_out,          // [B*NP]
                        unsigned long long* __restrict__ packed) {
    __shared__ float4 sbox[NOBJ];
    __shared__ float  sarea[NOBJ];
    __shared__ float  siou[NOBJ][257];   // +1 pad: avoid bank conflicts
    const int tid = threadIdx.x;
    const int b   = blockIdx.y;
    const int p0  = blockIdx.x * 256;
    const int p   = p0 + tid;

    if (tid < NOBJ) {
        float4 bx = boxes[b * NOBJ + tid];
        sbox[tid]  = bx;
        sarea[tid] = (bx.z - bx.x) * (bx.w - bx.y);
    }
    __syncthreads();

    if (p < NP) {
        float4 pr = priors[p];
        float ax0 = pr.x - 0.5f * pr.z, ay0 = pr.y - 0.5f * pr.w;
        float ax1 = pr.x + 0.5f * pr.z, ay1 = pr.y + 0.5f * pr.w;
        float pa  = (ax1 - ax0) * (ay1 - ay0);
        float best = -1.0f; int bi = 0;
        #pragma unroll 4
        for (int o = 0; o < NOBJ; ++o) {
            float4 bx = sbox[o];
            float ix = fminf(ax1, bx.z) - fmaxf(ax0, bx.x);
            float iy = fminf(ay1, bx.w) - fmaxf(ay0, bx.y);
            ix = fmaxf(ix, 0.0f); iy = fmaxf(iy, 0.0f);
            float inter = ix * iy;
            float iou = inter / (pa + sarea[o] - inter);
            siou[o][tid] = iou;
            if (iou > best) { best = iou; bi = o; }   // strict >: first argmax
        }
        ov_out[b * NP + p]  = best;
        obj_out[b * NP + p] = bi;
    } else {
        for (int o = 0; o < NOBJ; ++o) siou[o][tid] = -1.0f;
    }
    __syncthreads();

    if (tid < NOBJ) {
        float m = -1.0f; int mi = 0;
        for (int i = 0; i < 256; ++i) {
            float v = siou[tid][i];
            if (v > m) { m = v; mi = i; }
        }
        if (m >= 0.0f) {
            unsigned pb = (unsigned)(p0 + mi);
            unsigned long long key =
                ((unsigned long long)__float_as_uint(m) << 32) |
                (unsigned long long)(unsigned)(~pb);
            atomicMax(&packed[b * NOBJ + tid], key);
        }
    }
}

// ---------------------------------------------------------------------------
// Sequential per-image "force best prior of each valid object" (matches the
// reference's scatter with filtered-cumsum indices; last writer wins).
// ---------------------------------------------------------------------------
__global__ void k_force(const unsigned long long* __restrict__ packed,
                        float* __restrict__ ov, int* __restrict__ obj) {
    int b = threadIdx.x;
    if (b >= BATCH) return;
    int cnt = 0;
    for (int j = 0; j < NOBJ; ++j) {
        unsigned long long k = packed[b * NOBJ + j];
        float m    = __uint_as_float((unsigned)(k >> 32));
        unsigned p = ~(unsigned)(k & 0xffffffffu);
        if (m > 0.0f) {
            ov[b * NP + p]  = 1.0f;
            obj[b * NP + p] = cnt;
            cnt++;
        }
    }
}

__device__ __forceinline__ float smooth_l1(float d) {
    float a = fabsf(d);
    return (a < 1.0f) ? 0.5f * d * d : a - 0.5f;
}

// ---------------------------------------------------------------------------
// Per-prior: final class label, positive count, smooth-L1 localization loss.
// Block sums go through the WMMA reduction -> global f32 atomics.
// ---------------------------------------------------------------------------
__global__ void k_loc(const float4* __restrict__ pred_locs, // [B*NP]
                      const float4* __restrict__ boxes,     // [B*32]
                      const int*    __restrict__ labels,    // [B*32]
                      const float4* __restrict__ priors,    // [NP]
                      const float*  __restrict__ ov,
                      const int*    __restrict__ obj,
                      int*   __restrict__ lab_out,          // [B*NP]
                      float* __restrict__ accum) {          // [0]=conf [1]=loc [2]=npos
    const int b = blockIdx.y;
    const int p = blockIdx.x * 256 + threadIdx.x;
    float locp = 0.0f, np = 0.0f;
    if (p < NP) {
        float o  = ov[b * NP + p];
        int   oi = obj[b * NP + p];
        int  lab = labels[b * NOBJ + oi];
        if (o < THRESH)        lab = -1;
        if (o < THRESH - 0.1f) lab = 0;
        lab_out[b * NP + p] = lab;
        if (lab > 0) {
            np = 1.0f;
            float4 bx = boxes[b * NOBJ + oi];
            float bcx = 0.5f * (bx.x + bx.z), bcy = 0.5f * (bx.y + bx.w);
            float bw  = bx.z - bx.x,          bh  = bx.w - bx.y;
            float4 pr = priors[p];
            float g0 = (bcx - pr.x) * 10.0f / pr.z;
            float g1 = (bcy - pr.y) * 10.0f / pr.w;
            float g2 = __logf(bw / pr.z) * 5.0f;
            float g3 = __logf(bh / pr.w) * 5.0f;
            float4 pl = pred_locs[b * NP + p];
            locp = smooth_l1(pl.x - g0) + smooth_l1(pl.y - g1) +
                   smooth_l1(pl.z - g2) + smooth_l1(pl.w - g3);
        }
    }
    wave_sum_wmma_atomic(locp, &accum[1]);
    wave_sum_wmma_atomic(np,   &accum[2]);
}

// ---------------------------------------------------------------------------
// Sigmoid focal loss: coalesced float4 stream over all B*P*80 logits.
// logp = -log(1+e^-x), logn = logp - x, p = 1/(1+e^-x): 1 exp + 1 log / elem.
// ---------------------------------------------------------------------------
__global__ void k_focal(const float4* __restrict__ scores4,
                        const int* __restrict__ lab,
                        float* __restrict__ accum) {
    const unsigned total4 = (unsigned)BATCH * NP * (NCLS / 4);  // 7,211,520
    unsigned i      = blockIdx.x * 256u + threadIdx.x;
    unsigned stride = gridDim.x * 256u;
    float acc = 0.0f;
    for (; i < total4; i += stride) {
        __builtin_prefetch(&scores4[i + stride], 0, 0);  // global_prefetch
        unsigned prior = i / (NCLS / 4);
        int c0 = (int)(i - prior * (NCLS / 4)) * 4;
        int t  = lab[prior];
        float4 x4 = scores4[i];
        if (t >= 0) {
            float xs[4] = {x4.x, x4.y, x4.z, x4.w};
            #pragma unroll
            for (int j = 0; j < 4; ++j) {
                float x    = xs[j];
                int   c    = c0 + j + 1;              // 1-indexed class
                float enx  = __expf(-x);
                float pp   = 1.0f / (1.0f + enx);     // sigmoid(x)
                float logp = -__logf(1.0f + enx);     // log sigmoid(x)
                float logn = logp - x;                // log sigmoid(-x)
                float q    = 1.0f - pp;
                acc += (c == t) ? -ALPHA_F * q * q * logp
                                : -(1.0f - ALPHA_F) * pp * pp * logn;
            }
        }
    }
    wave_sum_wmma_atomic(acc, &accum[0]);
}

__global__ void k_final(const float* __restrict__ accum, float* __restrict__ out) {
    float n = fmaxf(accum[2], 1.0f);
    out[0] = accum[0] / n + accum[1] / (4.0f * n);
}

// ---------------------------------------------------------------------------
extern "C" void kernel_launch(void* const* d_in, const int* in_sizes, int n_in,
                              void* d_out, int out_size, void* d_ws, size_t ws_size,
                              hipStream_t stream) {
    const float4* pred_locs = (const float4*)d_in[0];
    const float4* scores4   = (const float4*)d_in[1];
    const float4* boxes     = (const float4*)d_in[2];
    const int*    labels    = (const int*)d_in[3];
    const float4* priors    = (const float4*)d_in[4];
    float* out = (float*)d_out;

    char* ws = (char*)d_ws;
    const size_t NBP = (size_t)BATCH * NP;
    float* ov   = (float*)(ws);
    int*   obj  = (int*)(ws + NBP * 4);
    int*   lab  = (int*)(ws + NBP * 8);
    unsigned long long* packed = (unsigned long long*)(ws + NBP * 12);
    float* accum = (float*)(ws + NBP * 12 + (size_t)BATCH * NOBJ * 8);

    const int pblocks = (NP + 255) / 256;   // 89

    k_init<<<1, 512, 0, stream>>>(packed, accum);
    k_match<<<dim3(pblocks, BATCH), 256, 0, stream>>>(boxes, priors, ov, obj, packed);
    k_force<<<1, BATCH, 0, stream>>>(packed, ov, obj);
    k_loc<<<dim3(pblocks, BATCH), 256, 0, stream>>>(pred_locs, boxes, labels, priors,
                                                    ov, obj, lab, accum);
    k_focal<<<2048, 256, 0, stream>>>(scores4, lab, accum);
    k_final<<<1, 1, 0, stream>>>(accum, out);
}